// PermutoEncoding_58188216926732
// MI455X (gfx1250) — compile-verified
//
#include <hip/hip_runtime.h>
#include <stdint.h>

#define POS_DIM   3
#define CAPACITY  262144       // 2^18
#define NR_LEVELS 16
#define NR_FEAT   2
#define N_POINTS  524288
#define HASH_MULT 2531011u

// round e to the nearest multiple of 4; ties -> down (matches reference
// "up - e < e - down ? up : down")
__device__ __forceinline__ float round_mult4(float e) {
    float v  = e * 0.25f;
    float up = ceilf(v)  * 4.0f;
    float dn = floorf(v) * 4.0f;
    return (up - e < e - dn) ? up : dn;
}

__device__ __forceinline__ uint32_t lds_addr(const void* p) {
    // generic pointers to LDS carry the 32-bit LDS offset in the low bits
    return (uint32_t)(uintptr_t)p;
}

__global__ __launch_bounds__(256) void permuto_fwd_kernel(
    const float* __restrict__ positions,   // (N, 3)
    const float* __restrict__ lattice,     // (L, C, 2)
    const float* __restrict__ scale,       // (L, 3)
    const float* __restrict__ shift,       // (L, 3)
    const float* __restrict__ anneal,      // (L,)
    float*       __restrict__ out)         // (N, L*2)
{
    // ---- async global->LDS staging (CDNA5) --------------------------------
    // scon: [0,48) scale, [48,96) shift, [96,112) anneal
    // spos: 16 points x 3 floats for this block
    __shared__ float scon[112];
    __shared__ float spos[48];

    const int t = threadIdx.x;
    if (t < 160) {
        const float* src;
        int off;
        uint32_t laddr;
        if (t < 48)       { src = scale;     off = t;                          laddr = lds_addr(&scon[t]); }
        else if (t < 96)  { src = shift;     off = t - 48;                     laddr = lds_addr(&scon[t]); }
        else if (t < 112) { src = anneal;    off = t - 96;                     laddr = lds_addr(&scon[t]); }
        else              { src = positions; off = blockIdx.x * 48 + (t-112);  laddr = lds_addr(&spos[t - 112]); }
        uint64_t gaddr = (uint64_t)(uintptr_t)(src + off);
        asm volatile("global_load_async_to_lds_b32 %0, %1, off"
                     :: "v"(laddr), "v"(gaddr) : "memory");
    }
    asm volatile("s_wait_asynccnt 0" ::: "memory");
    __syncthreads();

    // ---- one thread per (point, level); level minor for coalesced stores ----
    const int tid   = blockIdx.x * 256 + t;      // < N_POINTS * NR_LEVELS
    const int lpt   = t >> 4;                    // local point in block [0,16)
    const int level = t & 15;

    const float p0 = spos[lpt * 3 + 0];
    const float p1 = spos[lpt * 3 + 1];
    const float p2 = spos[lpt * 3 + 2];

    const float sc0 = scon[level * 3 + 0];
    const float sc1 = scon[level * 3 + 1];
    const float sc2 = scon[level * 3 + 2];
    const float sh0 = scon[48 + level * 3 + 0];
    const float sh1 = scon[48 + level * 3 + 1];
    const float sh2 = scon[48 + level * 3 + 2];
    const float ann = scon[96 + level];

    const float c0 = (p0 + sh0) * sc0;
    const float c1 = (p1 + sh1) * sc1;
    const float c2 = (p2 + sh2) * sc2;

    // elevation onto hyperplane H_3 (matches reference cumsum construction)
    const float t12 = c2 + c1;
    const float e0  = t12 + c0;
    const float e1  = t12 - c0;
    const float e2  = c2 - 2.0f * c1;
    const float e3  = -(3.0f * c2);

    // nearest remainder-0 lattice point
    float r0f = round_mult4(e0);
    float r1f = round_mult4(e1);
    float r2f = round_mult4(e2);
    float r3f = round_mult4(e3);

    const int sumq = (int)((r0f + r1f + r2f + r3f) * 0.25f);

    // descending rank of residuals, ties: earlier index ranks higher
    const float di0 = e0 - r0f;
    const float di1 = e1 - r1f;
    const float di2 = e2 - r2f;
    const float di3 = e3 - r3f;
    int k0 = 0, k1 = 0, k2 = 0, k3 = 0;
    if (di0 < di1) k0++; else k1++;
    if (di0 < di2) k0++; else k2++;
    if (di0 < di3) k0++; else k3++;
    if (di1 < di2) k1++; else k2++;
    if (di1 < di3) k1++; else k3++;
    if (di2 < di3) k2++; else k3++;

    k0 += sumq; k1 += sumq; k2 += sumq; k3 += sumq;
    if (k0 < 0) { k0 += 4; r0f += 4.0f; } else if (k0 > 3) { k0 -= 4; r0f -= 4.0f; }
    if (k1 < 0) { k1 += 4; r1f += 4.0f; } else if (k1 > 3) { k1 -= 4; r1f -= 4.0f; }
    if (k2 < 0) { k2 += 4; r2f += 4.0f; } else if (k2 > 3) { k2 -= 4; r2f -= 4.0f; }
    if (k3 < 0) { k3 += 4; r3f += 4.0f; } else if (k3 > 3) { k3 -= 4; r3f -= 4.0f; }

    // barycentric deltas
    const float d0 = (e0 - r0f) * 0.25f;
    const float d1 = (e1 - r1f) * 0.25f;
    const float d2 = (e2 - r2f) * 0.25f;
    const float d3 = (e3 - r3f) * 0.25f;

    // dd[k] = delta of the coord whose rank == k (ranks are a permutation of 0..3)
    const float dd0 = (k0 == 0) ? d0 : (k1 == 0) ? d1 : (k2 == 0) ? d2 : d3;
    const float dd1 = (k0 == 1) ? d0 : (k1 == 1) ? d1 : (k2 == 1) ? d2 : d3;
    const float dd2 = (k0 == 2) ? d0 : (k1 == 2) ? d1 : (k2 == 2) ? d2 : d3;
    const float dd3 = (k0 == 3) ? d0 : (k1 == 3) ? d1 : (k2 == 3) ? d2 : d3;

    // barycentric weights for vertices r = 0..3
    const float w0 = (dd3 + 1.0f) - dd0;
    const float w1 = dd2 - dd3;
    const float w2 = dd1 - dd2;
    const float w3 = dd0 - dd1;

    const int i0 = (int)r0f;
    const int i1 = (int)r1f;
    const int i2 = (int)r2f;

    const float2* __restrict__ table =
        (const float2*)lattice + ((size_t)level << 18);   // level * CAPACITY

    float a0 = 0.0f, a1 = 0.0f;
#pragma unroll
    for (int r = 0; r < 4; ++r) {
        const uint32_t key0 = (uint32_t)(i0 + r - ((k0 > 3 - r) ? 4 : 0));
        const uint32_t key1 = (uint32_t)(i1 + r - ((k1 > 3 - r) ? 4 : 0));
        const uint32_t key2 = (uint32_t)(i2 + r - ((k2 > 3 - r) ? 4 : 0));
        uint32_t h = key0 * HASH_MULT;
        h = (h + key1) * HASH_MULT;
        h = (h + key2) * HASH_MULT;
        const uint32_t idx = h & (CAPACITY - 1u);
        const float w = (r == 0) ? w0 : (r == 1) ? w1 : (r == 2) ? w2 : w3;
        const float2 val = table[idx];       // global_load_b64 gather (L2-resident)
        a0 = fmaf(w, val.x, a0);
        a1 = fmaf(w, val.y, a1);
    }

    float2 o;
    o.x = a0 * ann;
    o.y = a1 * ann;
    ((float2*)out)[tid] = o;                 // coalesced global_store_b64
}

extern "C" void kernel_launch(void* const* d_in, const int* in_sizes, int n_in,
                              void* d_out, int out_size, void* d_ws, size_t ws_size,
                              hipStream_t stream) {
    const float* positions = (const float*)d_in[0];   // (N,3)
    const float* lattice   = (const float*)d_in[1];   // (L,C,2)
    const float* scale     = (const float*)d_in[2];   // (L,3)
    const float* shift     = (const float*)d_in[3];   // (L,3)
    const float* anneal    = (const float*)d_in[4];   // (L,)
    float* out = (float*)d_out;

    const int total  = N_POINTS * NR_LEVELS;          // 8,388,608 threads
    const int block  = 256;                           // 8 wave32 waves
    const int grid   = total / block;                 // 32768 blocks
    permuto_fwd_kernel<<<grid, block, 0, stream>>>(positions, lattice, scale,
                                                   shift, anneal, out);
}